// PH_33062658244761
// MI455X (gfx1250) — compile-verified
//
#include <hip/hip_runtime.h>
#include <math.h>

#define BB 2
#define SS 512
#define HH 128
#define AA 7
#define MM (BB*SS)      // 1024
#define LN_EPS 1e-5f

typedef __attribute__((ext_vector_type(2))) float v2f;
typedef __attribute__((ext_vector_type(8))) float v8f;

// gfx1250 has native V_TANH_F32 (TRANS32). Prefer the builtin; fall back to
// inline asm (v_nop covers the TRANS-op result hazard), then to libm.
__device__ __forceinline__ float fast_tanh(float x) {
#if __has_builtin(__builtin_amdgcn_tanhf)
    return __builtin_amdgcn_tanhf(x);
#elif defined(__gfx1250__)
    float r;
    asm volatile("v_tanh_f32 %0, %1\n\tv_nop" : "=v"(r) : "v"(x));
    return r;
#else
    return tanhf(x);
#endif
}

__device__ __forceinline__ float gelu_tanh(float x) {
    float x3 = x * x * x;
    float t = fast_tanh(0.7978845608028654f * (x + 0.044715f * x3));
    return 0.5f * x * (1.0f + t);
}

// ---------------------------------------------------------------------------
// Kernel 1: five fused GEMMs  Y = X @ W + bias   (X: [1024,128], W: [128,128])
// f32 WMMA 16x16x4, one 16x16 tile per wave, K-loop of 32 WMMAs.
// grid = (64, 5); block = 256 (8 waves -> 8 tiles/block; 512 tiles/matrix)
// ---------------------------------------------------------------------------
__global__ void __launch_bounds__(256) gemm5_wmma(
    const float* __restrict__ X,
    const float* __restrict__ W0, const float* __restrict__ c0, float* __restrict__ Y0,
    const float* __restrict__ W1, const float* __restrict__ c1, float* __restrict__ Y1,
    const float* __restrict__ W2, const float* __restrict__ c2, float* __restrict__ Y2,
    const float* __restrict__ W3, const float* __restrict__ c3, float* __restrict__ Y3,
    const float* __restrict__ W4, const float* __restrict__ c4, float* __restrict__ Y4)
{
    const float* W; const float* bias; float* Y;
    switch (blockIdx.y) {
        case 0:  W = W0; bias = c0; Y = Y0; break;
        case 1:  W = W1; bias = c1; Y = Y1; break;
        case 2:  W = W2; bias = c2; Y = Y2; break;
        case 3:  W = W3; bias = c3; Y = Y3; break;
        default: W = W4; bias = c4; Y = Y4; break;
    }
    const int wave = threadIdx.x >> 5;
    const int lane = threadIdx.x & 31;
    const int tile = blockIdx.x * 8 + wave;   // 0..511
    const int tm = tile >> 3;                 // 64 M-tiles
    const int tn = tile & 7;                  // 8  N-tiles

    const int mRow  = tm * 16 + (lane & 15);  // A row for this lane
    const int nCol  = tn * 16 + (lane & 15);  // B col for this lane
    const int kHalf = (lane >> 4) << 1;       // lanes 16-31 carry K+2,K+3

    v8f acc = {};
    #pragma unroll 4
    for (int kb = 0; kb < HH; kb += 4) {
        const int k = kb + kHalf;
        v2f a, bf;
        a[0]  = X[mRow * HH + k];
        a[1]  = X[mRow * HH + k + 1];
        bf[0] = W[k * HH + nCol];
        bf[1] = W[(k + 1) * HH + nCol];
        acc = __builtin_amdgcn_wmma_f32_16x16x4_f32(
                  false, a, false, bf, (short)0, acc, false, false);
    }
    const float bv = bias[nCol];
    const int mBase = tm * 16 + ((lane >> 4) ? 8 : 0);
    #pragma unroll
    for (int r = 0; r < 8; ++r)
        Y[(mBase + r) * HH + nCol] = acc[r] + bv;
}

// ---------------------------------------------------------------------------
// Kernel 2: action head GeLU + LayerNorm(H).  grid = 1024 rows, block = 128
// ---------------------------------------------------------------------------
__global__ void __launch_bounds__(128) gelu_ln_k(
    const float* __restrict__ Yin, const float* __restrict__ g,
    const float* __restrict__ b,  float* __restrict__ Yout)
{
    __shared__ float s1[128];
    __shared__ float s2[128];
    const int m = blockIdx.x, h = threadIdx.x;
    const float y = gelu_tanh(Yin[m * HH + h]);
    s1[h] = y; s2[h] = y * y;
    __syncthreads();
    for (int off = 64; off > 0; off >>= 1) {
        if (h < off) { s1[h] += s1[h + off]; s2[h] += s2[h + off]; }
        __syncthreads();
    }
    const float mean = s1[0] * (1.0f / HH);
    const float var  = s2[0] * (1.0f / HH) - mean * mean;
    Yout[m * HH + h] = (y - mean) * rsqrtf(var + LN_EPS) * g[h] + b[h];
}

// ---------------------------------------------------------------------------
// Kernel 3: logits = actH @ aw2 + ab2   ([1024,128]x[128,7], trivial)
// ---------------------------------------------------------------------------
__global__ void __launch_bounds__(256) logits_k(
    const float* __restrict__ Xh, const float* __restrict__ W,
    const float* __restrict__ bias, float* __restrict__ Y)
{
    const int idx = blockIdx.x * 256 + threadIdx.x;
    if (idx >= MM * AA) return;
    const int m = idx / AA, a = idx % AA;
    float acc = bias[a];
    #pragma unroll 8
    for (int k = 0; k < HH; ++k) acc = fmaf(Xh[m * HH + k], W[k * AA + a], acc);
    Y[idx] = acc;
}

// ---------------------------------------------------------------------------
// Kernel 4: log_softmax over axis=1 (sequence).  grid = B*A = 14 columns.
// ---------------------------------------------------------------------------
__global__ void __launch_bounds__(256) logsoftmax_seq_k(
    const float* __restrict__ L, float* __restrict__ out)
{
    __shared__ float red[256];
    const int b = blockIdx.x / AA, a = blockIdx.x % AA;
    const int t = threadIdx.x;
    float mx = -3.402823466e38f;
    for (int s = t; s < SS; s += 256) mx = fmaxf(mx, L[(b * SS + s) * AA + a]);
    red[t] = mx; __syncthreads();
    for (int off = 128; off > 0; off >>= 1) {
        if (t < off) red[t] = fmaxf(red[t], red[t + off]);
        __syncthreads();
    }
    mx = red[0]; __syncthreads();
    float sum = 0.f;
    for (int s = t; s < SS; s += 256) sum += expf(L[(b * SS + s) * AA + a] - mx);
    red[t] = sum; __syncthreads();
    for (int off = 128; off > 0; off >>= 1) {
        if (t < off) red[t] += red[t + off];
        __syncthreads();
    }
    const float lse = mx + logf(red[0]);
    for (int s = t; s < SS; s += 256)
        out[(b * SS + s) * AA + a] = L[(b * SS + s) * AA + a] - lse;
}

// ---------------------------------------------------------------------------
// Kernel 5: pointer head. One block per (b,i) output row; one wave per (i,j)
// pair (lane l owns h = 4l..4l+3).  LN+projection fused algebraically:
//   score = rstd*(S3 - mean*C1) + C2,  C1 = sum(g*pw), C2 = sum(b*pw)+pb
// so the inner loop needs only three fused wave32 shuffle reductions.
// grid = (S, B), block = 256 (8 waves).
// ---------------------------------------------------------------------------
__global__ void __launch_bounds__(256) pointer_head_k(
    const float* __restrict__ hv, const float* __restrict__ hd,
    const float* __restrict__ g,  const float* __restrict__ bln,
    const float* __restrict__ pw, const float* __restrict__ pb,
    float* __restrict__ out)
{
    __shared__ float srow[SS];
    __shared__ float red[256];
    const int b = blockIdx.y, i = blockIdx.x;
    const int lane = threadIdx.x & 31;
    const int wave = threadIdx.x >> 5;
    const int h0 = lane * 4;

    const float4 hd4 = *(const float4*)(hd + (b * SS + i) * HH + h0);
    const float4 g4  = *(const float4*)(g + h0);
    const float4 b4  = *(const float4*)(bln + h0);
    const float4 pw4 = *(const float4*)(pw + h0);

    // per-lane g*pw, and block constants C1 = sum(g*pw), C2 = sum(b*pw)+pb
    const float gp0 = g4.x * pw4.x, gp1 = g4.y * pw4.y;
    const float gp2 = g4.z * pw4.z, gp3 = g4.w * pw4.w;
    float c1 = gp0 + gp1 + gp2 + gp3;
    float c2 = b4.x * pw4.x + b4.y * pw4.y + b4.z * pw4.z + b4.w * pw4.w;
    #pragma unroll
    for (int off = 16; off > 0; off >>= 1) {
        c1 += __shfl_xor(c1, off, 32);
        c2 += __shfl_xor(c2, off, 32);
    }
    const float C1 = c1;
    const float C2 = c2 + pb[0];

    for (int j = wave; j < SS; j += 8) {
        const float4 hv4 = *(const float4*)(hv + (b * SS + j) * HH + h0);
        const float y0 = gelu_tanh(hv4.x + hd4.x);
        const float y1 = gelu_tanh(hv4.y + hd4.y);
        const float y2 = gelu_tanh(hv4.z + hd4.z);
        const float y3 = gelu_tanh(hv4.w + hd4.w);
        float s1 = (y0 + y1) + (y2 + y3);
        float s2 = fmaf(y0, y0, fmaf(y1, y1, fmaf(y2, y2, y3 * y3)));
        float s3 = fmaf(y0, gp0, fmaf(y1, gp1, fmaf(y2, gp2, y3 * gp3)));
        #pragma unroll
        for (int off = 16; off > 0; off >>= 1) {
            s1 += __shfl_xor(s1, off, 32);
            s2 += __shfl_xor(s2, off, 32);
            s3 += __shfl_xor(s3, off, 32);
        }
        const float mean = s1 * (1.0f / HH);
        const float var  = s2 * (1.0f / HH) - mean * mean;
        const float rstd = rsqrtf(var + LN_EPS);
        if (lane == 0) srow[j] = rstd * (s3 - mean * C1) + C2;
    }
    __syncthreads();

    const int t = threadIdx.x;
    float mx = fmaxf(srow[t], srow[t + 256]);
    red[t] = mx; __syncthreads();
    for (int off = 128; off > 0; off >>= 1) {
        if (t < off) red[t] = fmaxf(red[t], red[t + off]);
        __syncthreads();
    }
    mx = red[0]; __syncthreads();
    float sm = expf(srow[t] - mx) + expf(srow[t + 256] - mx);
    red[t] = sm; __syncthreads();
    for (int off = 128; off > 0; off >>= 1) {
        if (t < off) red[t] += red[t + off];
        __syncthreads();
    }
    const float lse = mx + logf(red[0]);
    out[(b * SS + i) * SS + t]       = srow[t] - lse;
    out[(b * SS + i) * SS + t + 256] = srow[t + 256] - lse;
}

// ---------------------------------------------------------------------------
extern "C" void kernel_launch(void* const* d_in, const int* in_sizes, int n_in,
                              void* d_out, int out_size, void* d_ws, size_t ws_size,
                              hipStream_t stream) {
    const float* hiddens = (const float*)d_in[0];
    const float* aw1 = (const float*)d_in[1];  const float* ab1 = (const float*)d_in[2];
    const float* a_ln_g = (const float*)d_in[3]; const float* a_ln_b = (const float*)d_in[4];
    const float* aw2 = (const float*)d_in[5];  const float* ab2 = (const float*)d_in[6];
    const float* lhid_w = (const float*)d_in[7];  const float* lhid_b = (const float*)d_in[8];
    const float* lhead_w = (const float*)d_in[9]; const float* lhead_b = (const float*)d_in[10];
    const float* l_ln_g = (const float*)d_in[11]; const float* l_ln_b = (const float*)d_in[12];
    const float* l_proj_w = (const float*)d_in[13]; const float* l_proj_b = (const float*)d_in[14];
    const float* rhid_w = (const float*)d_in[15]; const float* rhid_b = (const float*)d_in[16];
    const float* rhead_w = (const float*)d_in[17]; const float* rhead_b = (const float*)d_in[18];
    const float* r_ln_g = (const float*)d_in[19]; const float* r_ln_b = (const float*)d_in[20];
    const float* r_proj_w = (const float*)d_in[21]; const float* r_proj_b = (const float*)d_in[22];

    float* ws = (float*)d_ws;
    float* act1   = ws;                // [M,H]
    float* actH   = ws + 1 * MM * HH;  // [M,H]
    float* lhv    = ws + 2 * MM * HH;
    float* lhd    = ws + 3 * MM * HH;
    float* rhv    = ws + 4 * MM * HH;
    float* rhd    = ws + 5 * MM * HH;
    float* logits = ws + 6 * MM * HH;  // [M,A]

    float* out_actions = (float*)d_out;                     // [B,S,A]
    float* out_lefts   = out_actions + MM * AA;             // [B,S,S]
    float* out_rights  = out_lefts + BB * SS * SS;          // [B,S,S]

    // 1) five projection GEMMs via f32 WMMA
    gemm5_wmma<<<dim3(64, 5), 256, 0, stream>>>(
        hiddens,
        aw1,     ab1,     act1,
        lhid_w,  lhid_b,  lhv,
        lhead_w, lhead_b, lhd,
        rhid_w,  rhid_b,  rhv,
        rhead_w, rhead_b, rhd);

    // 2) action head: GeLU + LN
    gelu_ln_k<<<MM, 128, 0, stream>>>(act1, a_ln_g, a_ln_b, actH);

    // 3) logits = actH @ aw2 + ab2
    logits_k<<<(MM * AA + 255) / 256, 256, 0, stream>>>(actH, aw2, ab2, logits);

    // 4) log_softmax over sequence axis
    logsoftmax_seq_k<<<BB * AA, 256, 0, stream>>>(logits, out_actions);

    // 5) pointer heads (dominant cost: B*S^2 pairwise GeLU+LN+dot + row softmax)
    pointer_head_k<<<dim3(SS, BB), 256, 0, stream>>>(
        lhv, lhd, l_ln_g, l_ln_b, l_proj_w, l_proj_b, out_lefts);
    pointer_head_k<<<dim3(SS, BB), 256, 0, stream>>>(
        rhv, rhd, r_ln_g, r_ln_b, r_proj_w, r_proj_b, out_rights);
}